// VFELayer_47227460386996
// MI455X (gfx1250) — compile-verified
//
#include <hip/hip_runtime.h>

typedef float v2f __attribute__((ext_vector_type(2)));
typedef float v4f __attribute__((ext_vector_type(4)));
typedef float v8f __attribute__((ext_vector_type(8)));

#define T_DIM 64
#define CIN   64
#define COUT  128
#define LDSW  66      // row stride (floats) in LDS: pad 64 -> 66 to kill bank conflicts
#define PTS   4       // points per block (W staged in LDS amortized 4x)

__global__ __launch_bounds__(256, 1) void vfe_wmma_kernel(
    const float* __restrict__ inp,   // [N, 64, 64]
    const float* __restrict__ Wmat,  // [128, 64]
    const float* __restrict__ gamma,
    const float* __restrict__ beta,
    const float* __restrict__ rmean,
    const float* __restrict__ rvar,
    float* __restrict__ out,         // [N, 64, 256]
    int npoints)
{
    __shared__ float sW[COUT * LDSW];   // 128 x 66 floats = 33.0 KB
    __shared__ float sX[T_DIM * LDSW];  //  64 x 66 floats = 16.5 KB
    __shared__ int   sMax[COUT];        // per-channel max (int-encoded nonneg float)

    const int tid    = threadIdx.x;
    const int lane   = tid & 31;
    const int wave   = tid >> 5;    // 0..7
    const int m_tile = wave & 3;    // which 16-row block of T
    const int n_grp  = wave >> 2;   // 0..1 -> channels [0,64) or [64,128)
    const int l16    = lane & 15;
    const int khalf  = (lane >> 4) * 2;   // 0 for lanes 0-15, 2 for lanes 16-31

    // ---- stage W into LDS once per block (rows padded to LDSW) ----
    for (int i = tid; i < COUT * (CIN / 4); i += 256) {
        int row  = i >> 4;          // / (CIN/4)
        int col4 = i & 15;
        v4f w4 = *(const v4f*)(Wmat + row * CIN + col4 * 4);
        *(v4f*)(&sW[row * LDSW + col4 * 4]) = w4;
    }

    // ---- per-lane BN constants for the 4 N-tiles this wave owns ----
    float scl[4], shf[4];
    {
        int cbase = n_grp * 64 + l16;
        #pragma unroll
        for (int j = 0; j < 4; ++j) {
            int c = cbase + j * 16;
            float s = gamma[c] * rsqrtf(rvar[c] + 1e-3f);
            scl[j] = s;
            shf[j] = beta[c] - rmean[c] * s;
        }
    }

    const int arow = m_tile * 16 + l16;   // LDS row of A fragment
    const int brow = n_grp * 64 + l16;    // LDS row (channel) of B fragment, tile 0

    for (int p = 0; p < PTS; ++p) {
        const int n = blockIdx.x * PTS + p;     // block-uniform
        if (n >= npoints) break;                // uniform branch: EXEC stays all-1s

        // init per-channel max + stage X[n] into LDS
        if (tid < COUT) sMax[tid] = 0;
        const float* Xg = inp + (size_t)n * (T_DIM * CIN);
        for (int i = tid; i < T_DIM * (CIN / 4); i += 256) {
            int row  = i >> 4;
            int col4 = i & 15;
            v4f x4 = *(const v4f*)(Xg + row * CIN + col4 * 4);
            *(v4f*)(&sX[row * LDSW + col4 * 4]) = x4;
        }
        __syncthreads();

        // ---- GEMM: C[16x16] += A[16x4] x B[4x16], K = 64 in steps of 4 ----
        v8f zero = {0.f, 0.f, 0.f, 0.f, 0.f, 0.f, 0.f, 0.f};
        v8f acc[4] = {zero, zero, zero, zero};

        #pragma unroll
        for (int k = 0; k < CIN; k += 4) {
            v2f a = *(const v2f*)(&sX[arow * LDSW + k + khalf]);
            #pragma unroll
            for (int j = 0; j < 4; ++j) {
                v2f b = *(const v2f*)(&sW[(brow + j * 16) * LDSW + k + khalf]);
                acc[j] = __builtin_amdgcn_wmma_f32_16x16x4_f32(
                    /*neg_a=*/false, a, /*neg_b=*/false, b,
                    /*c_mod=*/(short)0, acc[j],
                    /*reuse_a=*/false, /*reuse_b=*/false);
            }
        }

        // ---- fused BN (inference) + ReLU + per-channel max over T ----
        float* outp = out + (size_t)n * (T_DIM * 2 * COUT);
        float tmax[4];
        #pragma unroll
        for (int j = 0; j < 4; ++j) {
            float m = 0.0f;
            #pragma unroll
            for (int r = 0; r < 8; ++r) {
                float v = fmaxf(acc[j][r] * scl[j] + shf[j], 0.0f);
                acc[j][r] = v;
                m = fmaxf(m, v);
            }
            // fold rows r (lanes 0-15) with rows r+8 (lanes 16-31)
            m = fmaxf(m, __shfl_xor(m, 16, 32));
            tmax[j] = m;
        }
        if (lane < 16) {
            #pragma unroll
            for (int j = 0; j < 4; ++j) {
                int c = n_grp * 64 + j * 16 + lane;
                atomicMax(&sMax[c], __float_as_int(tmax[j]));  // valid: values >= 0
            }
        }

        // ---- store pointwise half: out[n, t, 0:128] ----
        #pragma unroll
        for (int j = 0; j < 4; ++j) {
            int c = n_grp * 64 + j * 16 + l16;
            #pragma unroll
            for (int r = 0; r < 8; ++r) {
                int trow = m_tile * 16 + (lane >> 4) * 8 + r;
                outp[(size_t)trow * (2 * COUT) + c] = acc[j][r];
            }
        }

        __syncthreads();   // sMax complete

        // ---- store repeated half: out[n, t, 128:256] = max over T ----
        for (int i = tid; i < T_DIM * COUT; i += 256) {
            int trow = i >> 7;       // / COUT
            int c    = i & (COUT - 1);
            outp[(size_t)trow * (2 * COUT) + COUT + c] = __int_as_float(sMax[c]);
        }

        __syncthreads();   // protect sX/sMax before next iteration overwrites
    }
}

extern "C" void kernel_launch(void* const* d_in, const int* in_sizes, int n_in,
                              void* d_out, int out_size, void* d_ws, size_t ws_size,
                              hipStream_t stream) {
    const float* inp   = (const float*)d_in[0];
    const float* Wmat  = (const float*)d_in[1];
    const float* gamma = (const float*)d_in[2];
    const float* beta  = (const float*)d_in[3];
    const float* rmean = (const float*)d_in[4];
    const float* rvar  = (const float*)d_in[5];
    float* out = (float*)d_out;

    int npoints = in_sizes[0] / (T_DIM * CIN);     // 20000
    int nblocks = (npoints + PTS - 1) / PTS;       // 5000
    vfe_wmma_kernel<<<nblocks, 256, 0, stream>>>(inp, Wmat, gamma, beta,
                                                 rmean, rvar, out, npoints);
}